// ParallelPairConstancyAxiom_46480136077414
// MI455X (gfx1250) — compile-verified
//
#include <hip/hip_runtime.h>
#include <hip/hip_bf16.h>
#include <math.h>

// ---------------------------------------------------------------------------
// ParallelPairConstancyAxiom on MI455X (gfx1250, wave32)
//
// dist[i,j] = |n_i . mid_j - off_i|  computed as 16x16 tiles with ONE
// V_WMMA_F32_16X16X4_F32 per tile (homogeneous K=3 coordinate folds the
// offset into the GEMM).  Quantiles (q1/median/q3) of the masked distance
// set are found with an exact two-level 16-bit radix select over the float
// bit patterns (valid since distances >= 0), avoiding the 64M-element sort.
// ---------------------------------------------------------------------------

typedef __attribute__((ext_vector_type(2))) float v2f;
typedef __attribute__((ext_vector_type(8))) float v8f;

#define PI_F 3.14159265358979323846f
#define ANGLE_TH 0.08726646259971647885f /* radians(5.0) */
#define TILES_PER_BLOCK 8               /* 8 waves (256 thr) = 8 j-tiles */

// ---- WMMA tile: returns raw dot products n_i.mid_j - off_i for a 16x16 tile.
// A (16x4 f32): lanes 0-15 hold {K0,K1} = {n_x, n_y}; lanes 16-31 hold
// {K2,K3} = {-off, 0}.  B (4x16 f32): lanes 0-15 hold {K0,K1} = {mid_x,
// mid_y}; lanes 16-31 hold {K2,K3} = {1, 0}.  (ISA 7.12.2 layouts.)
__device__ __forceinline__ v8f tile_dots(const float* nx, const float* ny,
                                         const float* offv, const float* midx,
                                         const float* midy, int i0, int j0,
                                         int lane) {
  const bool lohalf = lane < 16;
  const int l = lane & 15;
  v2f a, b;
  a.x = lohalf ? nx[i0 + l] : -offv[i0 + l];
  a.y = lohalf ? ny[i0 + l] : 0.0f;
  b.x = lohalf ? midx[j0 + l] : 1.0f;
  b.y = lohalf ? midy[j0 + l] : 0.0f;
  v8f c = {};
  c = __builtin_amdgcn_wmma_f32_16x16x4_f32(false, a, false, b, (short)0, c,
                                            false, false);
  return c;
}

__global__ void zero_u32_kernel(unsigned* p, long long n) {
  long long i = (long long)blockIdx.x * blockDim.x + threadIdx.x;
  if (i < n) p[i] = 0u;
}

// ---- Per-edge geometry: angle in [0,pi), normal, line offset, midpoint.
__global__ void edge_precompute_kernel(const float* __restrict__ pos,
                                       const int* __restrict__ src,
                                       const int* __restrict__ dst,
                                       float* angles, float* nx, float* ny,
                                       float* offv, float* midx, float* midy,
                                       int E) {
  int e = blockIdx.x * blockDim.x + threadIdx.x;
  if (e >= E) return;
  int s = src[e], d = dst[e];
  float px = pos[2 * s], py = pos[2 * s + 1];
  float qx = pos[2 * d], qy = pos[2 * d + 1];
  float dx = qx - px, dy = qy - py;
  float len = fmaxf(sqrtf(dx * dx + dy * dy), 1e-8f);
  float ux = dx / len, uy = dy / len;
  float a = atan2f(uy, ux);
  a = a - floorf(a / PI_F) * PI_F;  // mod pi, result in [0, pi)
  angles[e] = a;
  float nxv = -uy, nyv = ux;
  nx[e] = nxv;
  ny[e] = nyv;
  offv[e] = px * nxv + py * nyv;
  midx[e] = 0.5f * (px + qx);
  midy[e] = 0.5f * (py + qy);
}

// ---- Pass 1: count masked pairs + histogram of top 16 bits of dist bits.
__global__ void pass1_hist_kernel(const float* __restrict__ angles,
                                  const float* __restrict__ nx,
                                  const float* __restrict__ ny,
                                  const float* __restrict__ offv,
                                  const float* __restrict__ midx,
                                  const float* __restrict__ midy,
                                  unsigned* hist1, unsigned* meta, int E) {
  const int wave = threadIdx.x >> 5;
  const int lane = threadIdx.x & 31;
  const int tj = blockIdx.x * TILES_PER_BLOCK + wave;
  const int ti = blockIdx.y;
  unsigned cnt = 0;
  if (tj >= ti) {  // wave-uniform: below-diagonal tiles have no j>i pairs
    const int i0 = ti * 16, j0 = tj * 16;
    v8f c = tile_dots(nx, ny, offv, midx, midy, i0, j0, lane);
    const int n = j0 + (lane & 15);
    const float aJ = angles[n];
    const int mbase = i0 + ((lane < 16) ? 0 : 8);
#pragma unroll
    for (int r = 0; r < 8; ++r) {
      int m = mbase + r;
      float ad = fabsf(angles[m] - aJ);
      ad = fminf(ad, PI_F - ad);
      if ((ad < ANGLE_TH) && (n > m)) {
        unsigned bits = __float_as_uint(fabsf(c[r]));
        atomicAdd(&hist1[bits >> 16], 1u);
        ++cnt;
      }
    }
  }
  for (int k = 16; k >= 1; k >>= 1) cnt += __shfl_xor(cnt, k, 32);
  if (lane == 0 && cnt) atomicAdd(&meta[0], cnt);
}

// ---- Select the high-16 bin and in-bin rank for q1 / median / q3.
__global__ void scan1_kernel(const unsigned* __restrict__ hist1,
                             unsigned* meta) {
  unsigned n = meta[0];
  unsigned rk[3];
  rk[0] = (n >= 4) ? (n / 4 - 1) : 0;          // max(0, n//4 - 1)
  rk[1] = n / 2;                               // median rank
  unsigned rmax = n ? (n - 1) : 0;
  unsigned r3 = (3u * n) / 4;
  rk[2] = (r3 > rmax) ? rmax : r3;             // min(n-1, 3n//4)
  unsigned cum = 0;
  for (int b = 0; b < 65536; ++b) {
    unsigned c = hist1[b];
    for (int t = 0; t < 3; ++t)
      if (rk[t] >= cum && rk[t] < cum + c) {
        meta[1 + t] = (unsigned)b;
        meta[4 + t] = rk[t] - cum;
      }
    cum += c;
  }
}

// ---- Pass 2: low-16-bit histograms restricted to each target bin.
__global__ void pass2_hist_kernel(const float* __restrict__ angles,
                                  const float* __restrict__ nx,
                                  const float* __restrict__ ny,
                                  const float* __restrict__ offv,
                                  const float* __restrict__ midx,
                                  const float* __restrict__ midy,
                                  const unsigned* __restrict__ meta,
                                  unsigned* hist2, int E) {
  const int wave = threadIdx.x >> 5;
  const int lane = threadIdx.x & 31;
  const int tj = blockIdx.x * TILES_PER_BLOCK + wave;
  const int ti = blockIdx.y;
  if (tj < ti) return;  // wave-uniform skip
  const unsigned tb0 = meta[1], tb1 = meta[2], tb2 = meta[3];
  const int i0 = ti * 16, j0 = tj * 16;
  v8f c = tile_dots(nx, ny, offv, midx, midy, i0, j0, lane);
  const int n = j0 + (lane & 15);
  const float aJ = angles[n];
  const int mbase = i0 + ((lane < 16) ? 0 : 8);
#pragma unroll
  for (int r = 0; r < 8; ++r) {
    int m = mbase + r;
    float ad = fabsf(angles[m] - aJ);
    ad = fminf(ad, PI_F - ad);
    if ((ad < ANGLE_TH) && (n > m)) {
      unsigned bits = __float_as_uint(fabsf(c[r]));
      unsigned hi = bits >> 16, lo16 = bits & 0xFFFFu;
      if (hi == tb0) atomicAdd(&hist2[lo16], 1u);
      if (hi == tb1) atomicAdd(&hist2[65536 + lo16], 1u);
      if (hi == tb2) atomicAdd(&hist2[131072 + lo16], 1u);
    }
  }
}

// ---- Recover exact quantile values, compute mu and IQR margin.
__global__ void scan2_kernel(const unsigned* __restrict__ hist2,
                             const unsigned* __restrict__ meta, float* quant) {
  float q[3];
  for (int t = 0; t < 3; ++t) {
    unsigned bin = meta[1 + t], rk = meta[4 + t];
    unsigned cum = 0, low = 0;
    for (int b2 = 0; b2 < 65536; ++b2) {
      unsigned c = hist2[t * 65536 + b2];
      if (rk < cum + c) { low = (unsigned)b2; break; }
      cum += c;
    }
    q[t] = __uint_as_float((bin << 16) | low);
  }
  float iqr = fmaxf(q[2] - q[0], 1e-6f);
  quant[0] = q[1];         // mu (median)
  quant[1] = iqr * 0.75f;  // margin = iqr * 0.5 * IQR_SCALE(1.5)
}

// ---- Pass 3: per-pair loss, violation mask, deterministic partial sums.
__global__ void pass3_out_kernel(const float* __restrict__ angles,
                                 const float* __restrict__ nx,
                                 const float* __restrict__ ny,
                                 const float* __restrict__ offv,
                                 const float* __restrict__ midx,
                                 const float* __restrict__ midy,
                                 const float* __restrict__ quant,
                                 float* __restrict__ maskOut,
                                 float* __restrict__ plOut, float* partials,
                                 int E) {
  __shared__ float wsum[TILES_PER_BLOCK];
  const int wave = threadIdx.x >> 5;
  const int lane = threadIdx.x & 31;
  const int tj = blockIdx.x * TILES_PER_BLOCK + wave;
  const int ti = blockIdx.y;
  const int i0 = ti * 16, j0 = tj * 16;
  const int n = j0 + (lane & 15);
  const int mbase = i0 + ((lane < 16) ? 0 : 8);
  float sum = 0.f;
  if (tj < ti) {
    // Entire tile below diagonal: outputs are identically zero.
#pragma unroll
    for (int r = 0; r < 8; ++r) {
      size_t idx = (size_t)(mbase + r) * (size_t)E + (size_t)n;
      plOut[idx] = 0.f;
      maskOut[idx] = 0.f;
    }
  } else {
    v8f c = tile_dots(nx, ny, offv, midx, midy, i0, j0, lane);
    const float mu = quant[0], margin = quant[1];
    const float aJ = angles[n];
#pragma unroll
    for (int r = 0; r < 8; ++r) {
      int m = mbase + r;
      float ad = fabsf(angles[m] - aJ);
      ad = fminf(ad, PI_F - ad);
      bool msk = (ad < ANGLE_TH) && (n > m);
      float dist = fabsf(c[r]);
      float pl = msk ? fmaxf(fabsf(dist - mu) - margin, 0.f) : 0.f;
      size_t idx = (size_t)m * (size_t)E + (size_t)n;
      plOut[idx] = pl;
      maskOut[idx] = (pl > 0.f) ? 1.f : 0.f;
      sum += pl;
    }
  }
  for (int k = 16; k >= 1; k >>= 1) sum += __shfl_xor(sum, k, 32);
  if (lane == 0) wsum[wave] = sum;
  __syncthreads();
  if (threadIdx.x == 0) {
    float s = 0.f;
    for (int w = 0; w < TILES_PER_BLOCK; ++w) s += wsum[w];  // fixed order
    partials[(size_t)blockIdx.y * gridDim.x + blockIdx.x] = s;
  }
}

// ---- Deterministic final reduction; write loss and passthrough weight.
__global__ void finalize_kernel(const float* __restrict__ partials,
                                int numPartials,
                                const unsigned* __restrict__ meta,
                                const float* __restrict__ weight, float* out,
                                long long out_size) {
  __shared__ float red[256];
  float s = 0.f;
  for (int i = threadIdx.x; i < numPartials; i += 256) s += partials[i];
  red[threadIdx.x] = s;
  __syncthreads();
  for (int stride = 128; stride >= 1; stride >>= 1) {
    if ((int)threadIdx.x < stride) red[threadIdx.x] += red[threadIdx.x + stride];
    __syncthreads();
  }
  if (threadIdx.x == 0) {
    unsigned n = meta[0];
    float denom = (float)((n > 0u) ? n : 1u);
    out[0] = red[0] / denom;
    out[out_size - 1] = weight[0];
  }
}

extern "C" void kernel_launch(void* const* d_in, const int* in_sizes, int n_in,
                              void* d_out, int out_size, void* d_ws,
                              size_t ws_size, hipStream_t stream) {
  // Inputs: node_positions f32 (B*N*2), adjacency f32 (unused by reference),
  //         edge_index i32 (2,E), weight f32 (1).
  const float* pos = (const float*)d_in[0];
  const int* ei = (const int*)d_in[2];
  const float* weight = (const float*)d_in[3];
  const int E = in_sizes[2] / 2;  // 8192
  const int* src = ei;
  const int* dst = ei + E;

  // Workspace layout (floats/u32 words, contiguous).
  float* angles = (float*)d_ws;
  float* nx = angles + E;
  float* ny = nx + E;
  float* offv = ny + E;
  float* midx = offv + E;
  float* midy = midx + E;
  unsigned* hist1 = (unsigned*)(midy + E);
  unsigned* hist2 = hist1 + 65536;
  unsigned* meta = hist2 + 3 * 65536;  // [0]=n_pairs [1..3]=bin [4..6]=rank
  float* quant = (float*)(meta + 8);   // [0]=mu [1]=margin
  float* partials = quant + 8;

  const int tiles = E / 16;                    // 512
  const int gx = tiles / TILES_PER_BLOCK;      // 64
  const int numBlocks = gx * tiles;            // 32768 partial sums

  // Zero counters/histograms/partials (contiguous region after edge tables).
  long long zeroWords = 65536LL + 3LL * 65536 + 8 + 8 + numBlocks;
  zero_u32_kernel<<<(unsigned)((zeroWords + 255) / 256), 256, 0, stream>>>(
      hist1, zeroWords);

  edge_precompute_kernel<<<(E + 255) / 256, 256, 0, stream>>>(
      pos, src, dst, angles, nx, ny, offv, midx, midy, E);

  dim3 grid(gx, tiles);
  pass1_hist_kernel<<<grid, 256, 0, stream>>>(angles, nx, ny, offv, midx, midy,
                                              hist1, meta, E);
  scan1_kernel<<<1, 1, 0, stream>>>(hist1, meta);
  pass2_hist_kernel<<<grid, 256, 0, stream>>>(angles, nx, ny, offv, midx, midy,
                                              meta, hist2, E);
  scan2_kernel<<<1, 1, 0, stream>>>(hist2, meta, quant);

  float* outf = (float*)d_out;
  float* maskOut = outf + 1;
  float* plOut = maskOut + (size_t)E * (size_t)E;
  pass3_out_kernel<<<grid, 256, 0, stream>>>(angles, nx, ny, offv, midx, midy,
                                             quant, maskOut, plOut, partials,
                                             E);
  finalize_kernel<<<1, 256, 0, stream>>>(partials, numBlocks, meta, weight,
                                         outf, (long long)out_size);
}